// GraphEncoder_35287451304799
// MI455X (gfx1250) — compile-verified
//
#include <hip/hip_runtime.h>

#define DIM 128

typedef float v2f __attribute__((ext_vector_type(2)));
typedef float v8f __attribute__((ext_vector_type(8)));

// ---------------------------------------------------------------- utilities

__global__ __launch_bounds__(256) void zero_kernel(float* __restrict__ p, long n) {
    long i = (long)blockIdx.x * 256 + threadIdx.x;
    if (i < n) p[i] = 0.0f;
}

__global__ __launch_bounds__(256) void degree_kernel(const int* __restrict__ src,
                                                     const int* __restrict__ dst,
                                                     float* __restrict__ outdeg,
                                                     float* __restrict__ indeg,
                                                     int ne) {
    int e = blockIdx.x * 256 + threadIdx.x;
    if (e >= ne) return;
    unsafeAtomicAdd(&outdeg[src[e]], 1.0f);
    unsafeAtomicAdd(&indeg[dst[e]], 1.0f);
}

// in-place: deg -> rsqrt(max(deg,1))
__global__ __launch_bounds__(256) void norm_kernel(float* __restrict__ p, long n) {
    long i = (long)blockIdx.x * 256 + threadIdx.x;
    if (i < n) p[i] = __frsqrt_rn(fmaxf(p[i], 1.0f));
}

// ------------------------------------------------- fused GEMM + src_norm scale
// T[r, :] = (H[r, :] @ W) * srcn[r]
// One 256-thread block = 8 waves; W staged in LDS in pair-interleaved layout:
//   lW2[p*256 + (col ^ ((p&1)<<4))*2 + parity] = W[2p + parity][col]
// so a B fragment {W[kk][col], W[kk+1][col]} is ONE aligned ds_load_b64, and the
// XOR swizzle keeps the two lane-halves on disjoint LDS bank halves.
// Each wave computes a 16x128 strip: 8 col-tiles x 32 K-steps of
// V_WMMA_F32_16X16X4_F32 (full fp32 precision, matches reference).

__global__ __launch_bounds__(256) void gemm_scale_kernel(const float* __restrict__ H,
                                                         const float* __restrict__ W,
                                                         const float* __restrict__ srcn,
                                                         float* __restrict__ T,
                                                         int n) {
    __shared__ float lW2[DIM * DIM];   // 64 KB, pair-interleaved + swizzled
    for (int i = threadIdx.x; i < DIM * DIM; i += 256) {
        const int k = i >> 7;            // row of W
        const int c = i & (DIM - 1);     // col of W
        const int p = k >> 1;
        const int sc = c ^ ((p & 1) << 4);
        lW2[p * 256 + sc * 2 + (k & 1)] = W[i];
    }
    __syncthreads();

    const int wave = threadIdx.x >> 5;          // 0..7
    const int lane = threadIdx.x & 31;
    const int row0 = (blockIdx.x * 8 + wave) * 16;
    if (row0 >= n) return;                      // wave-uniform: EXEC stays all-1s

    const int half = lane >> 4;                 // 0: K pair (k0,k0+1), 1: (k0+2,k0+3)
    const int l15  = lane & 15;
    const int arow = row0 + l15;

    v8f acc[8];
#pragma unroll
    for (int j = 0; j < 8; ++j) acc[j] = (v8f)0.0f;

    const float* hrow = H + (size_t)arow * DIM;
#pragma unroll 4
    for (int k0 = 0; k0 < DIM; k0 += 4) {
        // A 16x4 fragment: lanes 0-15 -> (K=k0,k0+1); lanes 16-31 -> (K=k0+2,k0+3)
        v2f a = *(const v2f*)(hrow + k0 + 2 * half);
        const int p = (k0 >> 1) + half;         // pair index (p&1 == half)
        const float* bbase = lW2 + p * 256;
#pragma unroll
        for (int j = 0; j < 8; ++j) {
            const int col = j * 16 + l15;
            const int sc  = col ^ ((p & 1) << 4);
            v2f b = *(const v2f*)(bbase + sc * 2);   // single ds_load_b64
            acc[j] = __builtin_amdgcn_wmma_f32_16x16x4_f32(
                false, a, false, b, (short)0, acc[j], false, false);
        }
    }

    // epilogue: scale rows by src_norm and store (C layout: VGPR i -> row0+i+8*half)
#pragma unroll
    for (int i = 0; i < 8; ++i) {
        const int row = row0 + i + 8 * half;
        const float s = srcn[row];
        float* trow = T + (size_t)row * DIM + l15;
#pragma unroll
        for (int j = 0; j < 8; ++j) {
            trow[j * 16] = acc[j][i] * s;
        }
    }
}

// ------------------------------------------------- edge gather + scatter-add
// One wave per edge: 32 lanes x float4 = 128 features.
// t[src] is read as b128 (coalesced, L2-resident); hardware f32 atomics into agg[dst].

__global__ __launch_bounds__(256) void edge_scatter_kernel(const float* __restrict__ T,
                                                           const int* __restrict__ src,
                                                           const int* __restrict__ dst,
                                                           float* __restrict__ agg,
                                                           int ne) {
    long tid = (long)blockIdx.x * 256 + threadIdx.x;
    int e = (int)(tid >> 5);
    if (e >= ne) return;
    int lane = (int)(tid & 31);
    int s = src[e];
    int d = dst[e];
    const float4 v = ((const float4*)(T + (size_t)s * DIM))[lane];
    float* ap = agg + (size_t)d * DIM + lane * 4;
    unsafeAtomicAdd(ap + 0, v.x);
    unsafeAtomicAdd(ap + 1, v.y);
    unsafeAtomicAdd(ap + 2, v.z);
    unsafeAtomicAdd(ap + 3, v.w);
}

// ------------------------------------------------- dst_norm * agg + bias (+ReLU)

__global__ __launch_bounds__(256) void finalize_kernel(const float* __restrict__ agg,
                                                       const float* __restrict__ dstn,
                                                       const float* __restrict__ bias,
                                                       float* __restrict__ out,
                                                       long total, int relu) {
    long i = (long)blockIdx.x * 256 + threadIdx.x;
    if (i >= total) return;
    int row = (int)(i >> 7);        // /128
    int col = (int)(i & (DIM - 1));
    float v = agg[i] * dstn[row] + bias[col];
    out[i] = relu ? fmaxf(v, 0.0f) : v;
}

// ---------------------------------------------------------------- launcher

extern "C" void kernel_launch(void* const* d_in, const int* in_sizes, int n_in,
                              void* d_out, int out_size, void* d_ws, size_t ws_size,
                              hipStream_t stream) {
    const float* feats = (const float*)d_in[0];
    const float* W1    = (const float*)d_in[1];
    const float* b1    = (const float*)d_in[2];
    const float* W2    = (const float*)d_in[3];
    const float* b2    = (const float*)d_in[4];
    const float* W3    = (const float*)d_in[5];
    const float* b3    = (const float*)d_in[6];
    const int*   src   = (const int*)d_in[7];
    const int*   dst   = (const int*)d_in[8];

    const int  n   = in_sizes[0] / DIM;
    const int  ne  = in_sizes[7];
    const long nd  = (long)n * DIM;

    float* t    = (float*)d_ws;          // [n, 128] pre-scaled transformed feats
    float* agg  = t + nd;                // [n, 128] scatter accumulator
    float* h    = agg + nd;              // [n, 128] layer output
    float* srcn = h + nd;                // [n] out-degree -> rsqrt norm
    float* dstn = srcn + n;              // [n] in-degree  -> rsqrt norm

    const int B = 256;
    dim3 blk(B);
    dim3 gN((unsigned)((2L * n + B - 1) / B));          // both norm arrays
    dim3 gE((unsigned)((ne + B - 1) / B));
    dim3 gSc((unsigned)(((long)ne * 32 + B - 1) / B));
    dim3 gGemm((unsigned)((n + 127) / 128));
    dim3 gND((unsigned)((nd + B - 1) / B));

    // degrees -> norms (shared by all 3 layers)
    zero_kernel<<<gN, blk, 0, stream>>>(srcn, 2L * n);
    degree_kernel<<<gE, blk, 0, stream>>>(src, dst, srcn, dstn, ne);
    norm_kernel<<<gN, blk, 0, stream>>>(srcn, 2L * n);

    // layer 1
    gemm_scale_kernel<<<gGemm, blk, 0, stream>>>(feats, W1, srcn, t, n);
    zero_kernel<<<gND, blk, 0, stream>>>(agg, nd);
    edge_scatter_kernel<<<gSc, blk, 0, stream>>>(t, src, dst, agg, ne);
    finalize_kernel<<<gND, blk, 0, stream>>>(agg, dstn, b1, h, nd, 1);

    // layer 2
    gemm_scale_kernel<<<gGemm, blk, 0, stream>>>(h, W2, srcn, t, n);
    zero_kernel<<<gND, blk, 0, stream>>>(agg, nd);
    edge_scatter_kernel<<<gSc, blk, 0, stream>>>(t, src, dst, agg, ne);
    finalize_kernel<<<gND, blk, 0, stream>>>(agg, dstn, b2, h, nd, 1);

    // layer 3 (no ReLU) -> d_out
    gemm_scale_kernel<<<gGemm, blk, 0, stream>>>(h, W3, srcn, t, n);
    zero_kernel<<<gND, blk, 0, stream>>>(agg, nd);
    edge_scatter_kernel<<<gSc, blk, 0, stream>>>(t, src, dst, agg, ne);
    finalize_kernel<<<gND, blk, 0, stream>>>(agg, dstn, b3, (float*)d_out, nd, 0);
}